// TransducerLoss_21861383537343
// MI455X (gfx1250) — compile-verified
//
#include <hip/hip_runtime.h>
#include <hip/hip_bf16.h>
#include <stdint.h>

// Problem constants (from reference setup_inputs)
#define RB   4
#define RT   512
#define RU   100
#define RUP1 101
#define RV   1024
#define CH   8            // t-rows per TDM chunk (double buffered)

#define NEG_INF (-__builtin_inff())

// ---- TDM availability (device pass only; host pass falls back) ------------
#ifndef USE_TDM
#if defined(__has_builtin)
#if __has_builtin(__builtin_amdgcn_tensor_load_to_lds)
#define USE_TDM 1
#endif
#endif
#endif
#ifndef USE_TDM
#define USE_TDM 0
#endif

typedef unsigned int v4u __attribute__((ext_vector_type(4)));
typedef int          v8i __attribute__((ext_vector_type(8)));
typedef int          v4i __attribute__((ext_vector_type(4)));

// ---------------------------------------------------------------------------
// Stable log(exp(x)+exp(y)); exact passthrough when one side is -inf.
__device__ __forceinline__ float lae(float x, float y) {
    float m = fmaxf(x, y);
    if (m == NEG_INF) return NEG_INF;
    return m + log1pf(expf(-fabsf(x - y)));
}

// Inclusive add-scan over 128 threads (4 wave32s).
__device__ __forceinline__ float scan_add(float v, int lane, int wid, float* part) {
#pragma unroll
    for (int d = 1; d < 32; d <<= 1) {
        float n = __shfl_up(v, d, 32);
        if (lane >= d) v += n;
    }
    if (lane == 31) part[wid] = v;
    __syncthreads();
    float off = 0.0f;
    for (int w = 0; w < wid; ++w) off += part[w];
    __syncthreads();          // part[] reusable after this
    return v + off;
}

// Inclusive logcumsumexp-scan over 128 threads.
__device__ __forceinline__ float scan_lae(float v, int lane, int wid, float* part) {
#pragma unroll
    for (int d = 1; d < 32; d <<= 1) {
        float n = __shfl_up(v, d, 32);
        if (lane >= d) v = lae(v, n);
    }
    if (lane == 31) part[wid] = v;
    __syncthreads();
    float off = NEG_INF;
    for (int w = 0; w < wid; ++w) off = lae(off, part[w]);
    __syncthreads();
    return lae(v, off);
}

// ---------------------------------------------------------------------------
// TDM 2-D tile load: `rows` contiguous rows of `width` f32 (stride == width),
// global -> LDS. D# packing per CDNA5 ISA.
// Toolchain here is the clang-23 lane: 6-arg builtin
//   (uint32x4 g0, int32x8 g1, int32x4 g2, int32x4 g3, int32x8 pad, i32 cpol)
#if USE_TDM
__device__ __forceinline__ void tdm_load_tile(const float* gsrc, uint32_t lds_addr,
                                              uint32_t width, uint32_t rows) {
    uint64_t ga = (uint64_t)(uintptr_t)gsrc;
    v4u g0;
    g0[0] = 1u;                                       // count=1, user descriptor
    g0[1] = lds_addr;                                 // lds_addr [63:32]
    g0[2] = (uint32_t)ga;                             // global_addr low
    g0[3] = (uint32_t)((ga >> 32) & 0x01FFFFFFu)      // global_addr [56:32]
            | 0x80000000u;                            // type=2 ("image")
    v8i g1;
    g1[0] = (int)(2u << 16);                          // data_size=4B, wg_mask=0
    g1[1] = (int)((width & 0xFFFFu) << 16);           // tensor_dim0 lo16
    g1[2] = (int)(((width >> 16) & 0xFFFFu) |
                  ((rows & 0xFFFFu) << 16));          // dim0 hi16 | tensor_dim1 lo16
    g1[3] = (int)(((rows >> 16) & 0xFFFFu) |
                  ((width & 0xFFFFu) << 16));         // dim1 hi16 | tile_dim0
    g1[4] = (int)(rows & 0xFFFFu);                    // tile_dim1 (tile_dim2=0)
    g1[5] = (int)width;                               // tensor_dim0_stride lo32
    g1[6] = 0;                                        // stride hi16 | dim1_stride lo16
    g1[7] = 0;
    v4i gz4 = {0, 0, 0, 0};
    v8i gz8 = {0, 0, 0, 0, 0, 0, 0, 0};
    __builtin_amdgcn_tensor_load_to_lds(g0, g1, gz4, gz4, gz8, 0);
}
#endif

// Issue a chunk load (wave 0 only when TDM; cooperative copy otherwise).
__device__ __forceinline__ void chunk_load(const float* gsrc, float* ldst,
                                           int width, int rows, int tid) {
#if USE_TDM
    if (tid < 32)
        tdm_load_tile(gsrc, (uint32_t)(uintptr_t)ldst, (uint32_t)width, (uint32_t)rows);
#else
    int n = width * rows;
    for (int i = tid; i < n; i += 128) ldst[i] = gsrc[i];
#endif
}

__device__ __forceinline__ void chunk_wait(int more_pending, int tid) {
#if USE_TDM
    if (tid < 32) {
        if (more_pending) __builtin_amdgcn_s_wait_tensorcnt(2);  // oldest chunk done
        else              __builtin_amdgcn_s_wait_tensorcnt(0);
    }
#else
    (void)more_pending; (void)tid;
#endif
}

// ---------------------------------------------------------------------------
// Kernel 1: gather blank/emit from the huge logits tensor (2 of 1024 entries).
__global__ __launch_bounds__(256) void rnnt_gather(
        const float* __restrict__ logits, const int* __restrict__ labels,
        float* __restrict__ blank, float* __restrict__ emit) {
    int idx = blockIdx.x * 256 + threadIdx.x;       // idx = (b*T + t)*UP1 + u
    if (idx >= RB * RT * RUP1) return;
    int u  = idx % RUP1;
    int bt = idx / RUP1;                            // b*T + t
    const float* rowp = logits + (size_t)idx * RV;
    blank[idx] = rowp[0];
    if (u < RU) {
        int b = bt / RT;
        int lab = labels[b * RU + u];
        lab = lab < 0 ? 0 : (lab > RV - 1 ? RV - 1 : lab);
        emit[(size_t)bt * RU + u] = rowp[lab];
    }
}

// ---------------------------------------------------------------------------
// Kernel 2: per-batch alpha recursion. 1 block per batch, 128 threads.
// Thread u carries alpha row value row[u] in a register across t.
__global__ __launch_bounds__(128) void rnnt_dp(
        const float* __restrict__ blank, const float* __restrict__ emit,
        const int* __restrict__ tlen, const int* __restrict__ ulen,
        float* __restrict__ per) {
    __shared__ float sB[2][CH * RUP1];   // blank rows t-1
    __shared__ float sE[2][CH * RU];     // emit rows t
    __shared__ float partA[4];
    __shared__ float partL[4];

    int b    = blockIdx.x;
    int tid  = threadIdx.x;
    int lane = tid & 31;
    int wid  = tid >> 5;
    int u    = tid;
    bool act = (u < RUP1);

    int Tb = tlen[b]; Tb = Tb < 1 ? 1 : (Tb > RT ? RT : Tb);
    int Ub = ulen[b]; Ub = Ub < 1 ? 1 : (Ub > RU ? RU : Ub);

    const float* blk = blank + (size_t)b * RT * RUP1;
    const float* emt = emit  + (size_t)b * RT * RU;

    // row0[u] = sum_{j<u} emit[b,0,j]  (inclusive add-scan of shifted emit row)
    float e0  = (act && u >= 1) ? emt[u - 1] : 0.0f;
    float row = scan_add(e0, lane, wid, partA);

    int nIter = Tb - 1;                      // t = 1 .. Tb-1
    int nCh   = (nIter + CH - 1) / CH;

    if (nCh > 0) {                           // preload chunk 0
        int cnt0 = nIter < CH ? nIter : CH;
        chunk_load(blk,      &sB[0][0], RUP1, cnt0, tid);   // blank rows 0..
        chunk_load(emt + RU, &sE[0][0], RU,   cnt0, tid);   // emit rows 1..
    }

    for (int c = 0; c < nCh; ++c) {
        if (c + 1 < nCh) {                   // prefetch next chunk, then wait cur
            int t0n = 1 + (c + 1) * CH;
            int cnt = nIter - (c + 1) * CH; if (cnt > CH) cnt = CH;
            int nb  = (c + 1) & 1;
            chunk_load(blk + (size_t)(t0n - 1) * RUP1, &sB[nb][0], RUP1, cnt, tid);
            chunk_load(emt + (size_t)t0n * RU,         &sE[nb][0], RU,   cnt, tid);
            chunk_wait(1, tid);
        } else {
            chunk_wait(0, tid);
        }
        __syncthreads();                     // chunk c resident for all waves

        int buf = c & 1;
        int cnt = nIter - c * CH; if (cnt > CH) cnt = CH;
        for (int r = 0; r < cnt; ++r) {
            float bl = act               ? sB[buf][r * RUP1 + u]    : 0.0f;
            float ec = (act && u >= 1)   ? sE[buf][r * RU + (u - 1)] : 0.0f;
            float a  = row + bl;                               // prev + blank[t-1]
            float C  = scan_add(ec, lane, wid, partA);         // prefix-sum emit_t
            float x  = act ? (a - C) : NEG_INF;
            float L  = scan_lae(x, lane, wid, partL);          // logcumsumexp
            row = C + L;
        }
        __syncthreads();                     // buffer free before refill
    }

    if (act && u == Ub) per[b] = -row;
}

// ---------------------------------------------------------------------------
// Kernel 3: mean over batch (deterministic, single thread does the 4-term sum).
__global__ void rnnt_finalize(const float* __restrict__ per, float* __restrict__ out) {
    if (threadIdx.x == 0) {
        float s = 0.0f;
        for (int b = 0; b < RB; ++b) s += per[b];
        out[0] = s * (1.0f / RB);
    }
}

// ---------------------------------------------------------------------------
extern "C" void kernel_launch(void* const* d_in, const int* in_sizes, int n_in,
                              void* d_out, int out_size, void* d_ws, size_t ws_size,
                              hipStream_t stream) {
    (void)in_sizes; (void)n_in; (void)out_size; (void)ws_size;
    const float* logits = (const float*)d_in[0];
    const int*   labels = (const int*)d_in[1];
    const int*   tlen   = (const int*)d_in[2];
    const int*   ulen   = (const int*)d_in[3];

    float* blank = (float*)d_ws;                              // B*T*(U+1)
    float* emit  = blank + (size_t)RB * RT * RUP1;            // B*T*U
    float* per   = emit  + (size_t)RB * RT * RU;              // B

    int total = RB * RT * RUP1;
    rnnt_gather<<<(total + 255) / 256, 256, 0, stream>>>(logits, labels, blank, emit);
    rnnt_dp<<<RB, 128, 0, stream>>>(blank, emit, tlen, ulen, per);
    rnnt_finalize<<<1, 32, 0, stream>>>(per, (float*)d_out);
}